// canny_edge_27101243637979
// MI455X (gfx1250) — compile-verified
//
#include <hip/hip_runtime.h>
#include <math.h>
#include <stdint.h>

// ---------------------------------------------------------------------------
// Fused Canny+dilate for MI455X (gfx1250, wave32).
// One kernel, one pass: input read once (async global->LDS staging via
// GLOBAL_LOAD_ASYNC_TO_LDS_B32 + s_wait_asynccnt), all intermediates live in
// LDS, output written once. Memory-bound: ~134 MB compulsory traffic ->
// ~5.8 us floor at 23.3 TB/s; whole input fits in 192 MB L2 so tile halo
// re-reads are L2 hits.
// ---------------------------------------------------------------------------

#define TS   32            // output tile (TS x TS) per workgroup
#define GAW  44            // gray tile width  (TS + 12 halo)
#define GAP  45            // gray LDS row stride (odd -> no bank conflicts)
#define HBW  40            // h-blur cols (TS + 8)
#define HBP  41
#define BBP  41            // blurred 40x40
#define MGW  38            // mag tile (TS + 6)
#define MGP  39
#define NMW  36            // NMS tile (TS + 4)
#define NMP  37
#define RMP  33            // row-max stride (36 x 32 buffer)
#define NTH  256

#if defined(__has_builtin)
#  if __has_builtin(__builtin_amdgcn_global_load_async_to_lds_b32)
#    define HAVE_ASYNC_LDS 1
#  else
#    define HAVE_ASYNC_LDS 0
#  endif
#  if __has_builtin(__builtin_amdgcn_s_wait_asynccnt)
#    define HAVE_WAIT_ASYNC 1
#  else
#    define HAVE_WAIT_ASYNC 0
#  endif
#else
#  define HAVE_ASYNC_LDS 0
#  define HAVE_WAIT_ASYNC 0
#endif

#if !HAVE_ASYNC_LDS
#  warning "gfx1250 async global->LDS builtin NOT found; using VGPR staging fallback"
#endif
#if !HAVE_WAIT_ASYNC
#  warning "s_wait_asynccnt builtin NOT found; using inline asm"
#endif

typedef __attribute__((address_space(1))) int as1_int;
typedef __attribute__((address_space(3))) int as3_int;

// jnp.pad 'reflect' with pad<=2 on a 512 axis: -1->1, -2->2, 512->510, 513->509
__device__ __forceinline__ int reflect512(int i) {
  i = (i < 0) ? -i : i;
  return (i > 511) ? (1022 - i) : i;
}

__global__ __launch_bounds__(NTH) void canny_dilate_fused(
    const float* __restrict__ xin, float* __restrict__ out) {
  // --- LDS pipeline buffers -------------------------------------------------
#if HAVE_ASYNC_LDS
  __shared__ float raw0[GAW * GAP];
  __shared__ float raw1[GAW * GAP];
  __shared__ float raw2[GAW * GAP];
#endif
  __shared__ float gA[GAW * GAP];    // gray, reflect-padded coords [t-6, t+37]
  __shared__ float hB[GAW * HBP];    // horizontal blur, rows [t-6,t+37], cols [t-4,t+35]
  __shared__ float bB[HBW * BBP];    // blurred, [t-4, t+35]^2
  __shared__ float magL[MGW * MGP];  // |grad|, [t-3, t+34]^2 (0 outside image)
  __shared__ int   angL[MGW * MGP];  // quantized direction index
  __shared__ float nmL[NMW * NMP];   // NMS'd mag, [t-2, t+33]^2 (-1e4 outside image)
  __shared__ float rmL[NMW * RMP];   // separable dilation: row max

  const int tid = threadIdx.x;
  const int tx0 = blockIdx.x * TS;
  const int ty0 = blockIdx.y * TS;
  const int b   = blockIdx.z;
  const size_t plane = 512u * 512u;
  const float* base = xin + (size_t)b * 3u * plane;

  // ===== Stage 1: async-stage RGB tiles into LDS, fuse gray conversion =====
  // gray = 0.299*ch2 + 0.587*ch1 + 0.114*ch0 (after x[:, ::-1])
#if HAVE_ASYNC_LDS
  for (int p = tid; p < GAW * GAW; p += NTH) {
    const int r = p / GAW, c = p % GAW;
    const int gy = reflect512(ty0 - 6 + r);
    const int gx = reflect512(tx0 - 6 + c);
    const size_t off = (size_t)gy * 512 + gx;
    const int l = r * GAP + c;
    __builtin_amdgcn_global_load_async_to_lds_b32(
        (as1_int*)(base + off),             (as3_int*)&raw0[l], 0, 0);
    __builtin_amdgcn_global_load_async_to_lds_b32(
        (as1_int*)(base + plane + off),     (as3_int*)&raw1[l], 0, 0);
    __builtin_amdgcn_global_load_async_to_lds_b32(
        (as1_int*)(base + 2 * plane + off), (as3_int*)&raw2[l], 0, 0);
  }
#if HAVE_WAIT_ASYNC
  __builtin_amdgcn_s_wait_asynccnt(0);
#else
  asm volatile("s_wait_asynccnt 0x0" ::: "memory");
#endif
  __syncthreads();
  for (int p = tid; p < GAW * GAW; p += NTH) {
    const int r = p / GAW, c = p % GAW;
    const int l = r * GAP + c;
    gA[l] = 0.114f * raw0[l] + 0.587f * raw1[l] + 0.299f * raw2[l];
  }
#else
  for (int p = tid; p < GAW * GAW; p += NTH) {
    const int r = p / GAW, c = p % GAW;
    const int gy = reflect512(ty0 - 6 + r);
    const int gx = reflect512(tx0 - 6 + c);
    const size_t off = (size_t)gy * 512 + gx;
    const float c0 = base[off];
    const float c1 = base[plane + off];
    const float c2 = base[2 * plane + off];
    gA[r * GAP + c] = 0.114f * c0 + 0.587f * c1 + 0.299f * c2;
  }
#endif
  __syncthreads();

  // ===== Stage 2: horizontal 5-tap Gaussian (sigma=1) ======================
  const float w0 = 0.054488685f, w1 = 0.24420135f, w2 = 0.40261996f;
  for (int p = tid; p < GAW * HBW; p += NTH) {
    const int r = p / HBW, c = p % HBW;
    const float* g = &gA[r * GAP + c];
    hB[r * HBP + c] = w0 * (g[0] + g[4]) + w1 * (g[1] + g[3]) + w2 * g[2];
  }
  __syncthreads();

  // ===== Stage 3: vertical 5-tap Gaussian ==================================
  for (int p = tid; p < HBW * HBW; p += NTH) {
    const int r = p / HBW, c = p % HBW;
    bB[r * BBP + c] =
        w0 * (hB[(r)     * HBP + c] + hB[(r + 4) * HBP + c]) +
        w1 * (hB[(r + 1) * HBP + c] + hB[(r + 3) * HBP + c]) +
        w2 *  hB[(r + 2) * HBP + c];
  }
  __syncthreads();

  // ===== Stage 4: Sobel (edge-clamped), magnitude, octant ==================
  for (int p = tid; p < MGW * MGW; p += NTH) {
    const int r = p / MGW, c = p % MGW;
    const int y = ty0 - 3 + r, xx = tx0 - 3 + c;
    float m = 0.0f;
    int idx = 0;
    if ((unsigned)y < 512u && (unsigned)xx < 512u) {
      const int ym = max(y - 1, 0)    - (ty0 - 4);
      const int y0 = y                - (ty0 - 4);
      const int yp = min(y + 1, 511)  - (ty0 - 4);
      const int xm = max(xx - 1, 0)   - (tx0 - 4);
      const int x0 = xx               - (tx0 - 4);
      const int xp = min(xx + 1, 511) - (tx0 - 4);
      const float a00 = bB[ym * BBP + xm], a01 = bB[ym * BBP + x0], a02 = bB[ym * BBP + xp];
      const float a10 = bB[y0 * BBP + xm],                           a12 = bB[y0 * BBP + xp];
      const float a20 = bB[yp * BBP + xm], a21 = bB[yp * BBP + x0], a22 = bB[yp * BBP + xp];
      const float gxv = 0.125f * ((a02 - a00) + (a22 - a20)) + 0.25f * (a12 - a10);
      const float gyv = 0.125f * ((a20 - a00) + (a22 - a02)) + 0.25f * (a21 - a01);
      m = sqrtf(gxv * gxv + gyv * gyv + 1e-6f);
      // Octant of atan2(gy,gx) rounded to 45 deg, mod 8 — done with two
      // slope comparisons instead of a libm atan2 (equivalent away from the
      // measure-zero 22.5+45k deg boundaries).
      const float ax = fabsf(gxv), ay = fabsf(gyv);
      const float T1 = 0.41421356f;  // tan(22.5 deg)
      const float T2 = 2.41421356f;  // tan(67.5 deg)
      if (ay <= T1 * ax)      idx = (gxv >= 0.0f) ? 0 : 4;
      else if (ay >= T2 * ax) idx = (gyv >= 0.0f) ? 2 : 6;
      else if (gyv >= 0.0f)   idx = (gxv >= 0.0f) ? 1 : 3;
      else                    idx = (gxv >= 0.0f) ? 7 : 5;
    }
    magL[r * MGP + c] = m;   // 0 outside image == zero-pad for NMS conv
    angL[r * MGP + c] = idx;
  }
  __syncthreads();

  // ===== Stage 5: directional NMS + threshold ==============================
  // OFFS dy: {0,1,1,1,0,-1,-1,-1}  dx: {1,1,0,-1,-1,-1,0,1}, 4-bit nibble LUT
  for (int p = tid; p < NMW * NMW; p += NTH) {
    const int r = p / NMW, c = p % NMW;
    const int y = ty0 - 2 + r, xx = tx0 - 2 + c;
    float v = -10000.0f;  // constant pad value for the dilation stage
    if ((unsigned)y < 512u && (unsigned)xx < 512u) {
      const int lr = r + 1, lc = c + 1;
      const float m = magL[lr * MGP + lc];
      const int idx = angL[lr * MGP + lc];
      const int dy = (int)(((0xFFF01110u >> (idx * 4)) & 0xFu) ^ 8u) - 8;
      const int dx = (int)(((0x10FFF011u >> (idx * 4)) & 0xFu) ^ 8u) - 8;
      const float pos = m - magL[(lr + dy) * MGP + (lc + dx)];
      const float neg = m - magL[(lr - dy) * MGP + (lc - dx)];
      v = (fminf(pos, neg) > 0.0f) ? m : 0.0f;
    }
    nmL[r * NMP + c] = v;
  }
  __syncthreads();

  // ===== Stage 6: separable 5x5 max-dilation ===============================
  for (int p = tid; p < NMW * TS; p += NTH) {  // 36 rows x 32 cols row-max
    const int r = p / TS, c = p % TS;
    const float* n = &nmL[r * NMP + c];
    float mx = fmaxf(fmaxf(fmaxf(n[0], n[1]), fmaxf(n[2], n[3])), n[4]);
    rmL[r * RMP + c] = mx;
  }
  __syncthreads();

  float* outb = out + (size_t)b * plane;
  for (int p = tid; p < TS * TS; p += NTH) {
    const int r = p / TS, c = p % TS;
    float mx = rmL[r * RMP + c];
    mx = fmaxf(mx, rmL[(r + 1) * RMP + c]);
    mx = fmaxf(mx, rmL[(r + 2) * RMP + c]);
    mx = fmaxf(mx, rmL[(r + 3) * RMP + c]);
    mx = fmaxf(mx, rmL[(r + 4) * RMP + c]);
    outb[(size_t)(ty0 + r) * 512 + (tx0 + c)] = mx;
  }
}

extern "C" void kernel_launch(void* const* d_in, const int* in_sizes, int n_in,
                              void* d_out, int out_size, void* d_ws, size_t ws_size,
                              hipStream_t stream) {
  (void)in_sizes; (void)n_in; (void)d_ws; (void)ws_size; (void)out_size;
  const float* x = (const float*)d_in[0];
  float* out = (float*)d_out;
  dim3 grid(512 / TS, 512 / TS, 32);
  canny_dilate_fused<<<grid, NTH, 0, stream>>>(x, out);
}